// BoxDetectionLoss_34411277975620
// MI455X (gfx1250) — compile-verified
//
#include <hip/hip_runtime.h>
#include <hip/hip_bf16.h>

// Problem constants (fixed by the reference: B=32, C=12, H=W=256, T=64)
constexpr int kB = 32;
constexpr int kH = 256;
constexpr int kW = 256;
constexpr int kT = 64;
constexpr int kHW = kH * kW;                 // 65536 floats per channel map
constexpr int kNElem = kB * 3 * kHW;         // 6,291,456 v_raw elements
constexpr int kNF4 = kNElem / 4;             // 1,572,864 float4 loads
constexpr float kInvN = 1.0f / (float)kNElem;

typedef __attribute__((ext_vector_type(2))) float v2f;
typedef __attribute__((ext_vector_type(8))) float v8f;

__device__ __forceinline__ float softplus_f(float x) {
  // numerically stable log(1+e^x)
  return fmaxf(x, 0.0f) + log1pf(expf(-fabsf(x)));
}

__device__ __forceinline__ float sigmoid_f(float x) {
  return 1.0f / (1.0f + expf(-x));
}

// Exact 32-lane sum. Stage 1 uses V_WMMA_F32_16X16X4_F32 with A=ones,
// B=(acc,0): D[m][n] = B[0][n] + B[2][n] = acc(n) + acc(n+16), so d[0]
// holds the (lane, lane+16) pairwise sum, replicated across both halves.
// Stage 2 folds the 16 distinct values with xor-shuffles. All adds are
// f32 round-to-nearest-even -> bit-compatible with a plain tree sum.
__device__ __forceinline__ float wave_sum32(float acc) {
#if __has_builtin(__builtin_amdgcn_wmma_f32_16x16x4_f32)
  v2f a; a[0] = 1.0f; a[1] = 1.0f;      // A (16x4) = all ones
  v2f b; b[0] = acc;  b[1] = 0.0f;      // B rows 0/2 = lane partials, rows 1/3 = 0
  v8f c = {};
  v8f d = __builtin_amdgcn_wmma_f32_16x16x4_f32(
      /*neg_a=*/false, a, /*neg_b=*/false, b,
      /*c_mod=*/(short)0, c, /*reuse_a=*/false, /*reuse_b=*/false);
  float t = d[0];                       // acc(lane&15) + acc((lane&15)+16)
#else
  float t = acc + __shfl_xor(acc, 16, 32);
#endif
  t += __shfl_xor(t, 8, 32);
  t += __shfl_xor(t, 4, 32);
  t += __shfl_xor(t, 2, 32);
  t += __shfl_xor(t, 1, 32);
  return t;                              // total in every lane
}

// ---------------------------------------------------------------------------
// Kernel 1: sum softplus(v_raw) over all (b, anchor, h, w).
// v_raw = policy[b, 4k+2, :, :]  (channel layout after reshape (B,3,4,H,W)).
// Pure streaming reduction: 25.2 MB read, HBM-bound (~1.1 us at 23.3 TB/s).
// ---------------------------------------------------------------------------
__global__ void validity_kernel(const float* __restrict__ policy,
                                float* __restrict__ acc_ws) {
  float acc = 0.0f;
  int tid = blockIdx.x * blockDim.x + threadIdx.x;
  int stride = gridDim.x * blockDim.x;

  for (int i = tid; i < kNF4; i += stride) {
    int map = i >> 14;                   // which of the 96 v-channel maps
    int off4 = i & 16383;                // float4 offset inside the map
    int b = map / 3;
    int k = map - b * 3;
    const float4* src =
        reinterpret_cast<const float4*>(policy + (size_t)(b * 12 + k * 4 + 2) * kHW) + off4;

    // Prefetch next grid-stride iteration (emits global_prefetch_b8).
    int j = i + stride;
    if (j < kNF4) {
      int map2 = j >> 14;
      int b2 = map2 / 3;
      int k2 = map2 - b2 * 3;
      const float* nxt = policy + (size_t)(b2 * 12 + k2 * 4 + 2) * kHW + (size_t)(j & 16383) * 4;
      __builtin_prefetch(nxt, 0, 0);
    }

    float4 v = *src;
    acc += softplus_f(v.x) + softplus_f(v.y) + softplus_f(v.z) + softplus_f(v.w);
  }

  // Wave reduce (WMMA + shuffles), then one atomic per block.
  __shared__ float sred[8];
  float w = wave_sum32(acc);
  int lane = threadIdx.x & 31;
  int wid = threadIdx.x >> 5;
  if (lane == 0) sred[wid] = w;
  __syncthreads();
  if (threadIdx.x == 0) {
    float s = 0.0f;
    int nw = blockDim.x >> 5;
    for (int i = 0; i < nw; ++i) s += sred[i];
    atomicAdd(acc_ws, s);
  }
}

// ---------------------------------------------------------------------------
// Kernel 2: sparse matched corrections. One block per batch, one thread per
// target. A target contributes iff (a) mask, (b) no earlier target with an
// identical box (first-match-wins dedup), (c) at anchor k the rounded
// prediction at pixel (tr1,tc1) equals (tr2,tc2). Matched correction:
//   -v_raw  +  (sigmoid(p)-prob)^2  +  |pred_r2-tr2| + |pred_c2-tc2|
// (-v_raw is the exact delta softplus(-v) - softplus(v)).
// ---------------------------------------------------------------------------
__global__ void correction_kernel(const float* __restrict__ policy,
                                  const float* __restrict__ probs,
                                  const int* __restrict__ boxes,
                                  const unsigned char* __restrict__ mask,
                                  float* __restrict__ acc_ws) {
  __shared__ int sbox[kT][4];
  __shared__ unsigned char smask[kT];

  int b = blockIdx.x;
  int t = threadIdx.x;
  const int* my = boxes + (size_t)(b * kT + t) * 4;
  int r1 = my[0], c1 = my[1], r2 = my[2], c2 = my[3];
  unsigned char mk = mask[b * kT + t];
  sbox[t][0] = r1; sbox[t][1] = c1; sbox[t][2] = r2; sbox[t][3] = c2;
  smask[t] = mk;
  __syncthreads();

  bool dup = false;
  for (int u = 0; u < t; ++u) {
    if (smask[u] && sbox[u][0] == r1 && sbox[u][1] == c1 &&
        sbox[u][2] == r2 && sbox[u][3] == c2) { dup = true; break; }
  }

  float corr = 0.0f;
  if (mk && !dup) {
    float prob = probs[b * kT + t];
    float fr2 = (float)r2, fc2 = (float)c2;
    for (int k = 0; k < 3; ++k) {
      size_t base = (size_t)(b * 12 + k * 4) * kHW + (size_t)r1 * kW + (size_t)c1;
      float dr = policy[base];
      float dc = policy[base + kHW];
      float vr = policy[base + 2 * (size_t)kHW];
      float pr = policy[base + 3 * (size_t)kHW];

      float pred_r2 = fminf(fmaxf((float)r1 + sigmoid_f(dr) * 9.0f, 0.0f), (float)(kH - 1));
      float pred_c2 = fminf(fmaxf((float)c1 + sigmoid_f(dc) * 16.0f, 0.0f), (float)(kW - 1));
      int ri = (int)rintf(pred_r2);      // round-half-even, matches jnp.round
      int ci = (int)rintf(pred_c2);

      if (ri == r2 && ci == c2) {
        float sp = sigmoid_f(pr);
        float dpref = sp - prob;
        corr += -vr + dpref * dpref + fabsf(pred_r2 - fr2) + fabsf(pred_c2 - fc2);
      }
    }
  }
  if (corr != 0.0f) atomicAdd(acc_ws, corr);
}

__global__ void zero_kernel(float* __restrict__ ws) { ws[0] = 0.0f; }

__global__ void finalize_kernel(const float* __restrict__ ws, float* __restrict__ out) {
  out[0] = ws[0] * kInvN;
}

extern "C" void kernel_launch(void* const* d_in, const int* in_sizes, int n_in,
                              void* d_out, int out_size, void* d_ws, size_t ws_size,
                              hipStream_t stream) {
  const float* policy = (const float*)d_in[0];            // (32,12,256,256) f32
  const float* probs = (const float*)d_in[1];             // (32,64) f32
  const int* boxes = (const int*)d_in[2];                 // (32,64,4) i32
  const unsigned char* mask = (const unsigned char*)d_in[3]; // (32,64) bool8
  float* out = (float*)d_out;                             // scalar f32
  float* ws = (float*)d_ws;                               // [0] = accumulator

  zero_kernel<<<1, 1, 0, stream>>>(ws);
  validity_kernel<<<768, 256, 0, stream>>>(policy, ws);   // 196608 lanes, 8 f4/lane
  correction_kernel<<<kB, kT, 0, stream>>>(policy, probs, boxes, mask, ws);
  finalize_kernel<<<1, 1, 0, stream>>>(ws, out);
}